// TripletHardLoss_74801150427888
// MI455X (gfx1250) — compile-verified
//
#include <hip/hip_runtime.h>
#include <math.h>

#define EPS    1e-6f
#define MARGIN 1.0f
#define BROWS  512
#define DIM    256

typedef __attribute__((ext_vector_type(2))) float v2f;
typedef __attribute__((ext_vector_type(8))) float v8f;

// ---------------------------------------------------------------------------
// Kernel 1: per-row stats. One wave32 per row.
//   na[i]  = ||a_i||^2
//   nb[i]  = ||b_i - eps||^2
//   dap[i] = ||a_i - b_i + eps||          (anchor-positive distance)
//   cd[i]  = relu(dap - ||a-c+eps|| + M) + relu(dap - ||a-d+eps|| + M)
// ---------------------------------------------------------------------------
__global__ void rowstats_kernel(const float* __restrict__ a,
                                const float* __restrict__ b,
                                const float* __restrict__ c,
                                const float* __restrict__ d,
                                float* __restrict__ na,
                                float* __restrict__ nb,
                                float* __restrict__ dap,
                                float* __restrict__ cd) {
    const int lane = threadIdx.x & 31;
    const int wave = threadIdx.x >> 5;
    const int row  = blockIdx.x * (blockDim.x >> 5) + wave;
    if (row >= BROWS) return;

    const size_t base = (size_t)row * DIM;
    float sa = 0.f, sb = 0.f, sab = 0.f, sac = 0.f, sad = 0.f;
    #pragma unroll
    for (int e = lane; e < DIM; e += 32) {
        const float av = a[base + e];
        const float bv = b[base + e];
        const float cv = c[base + e];
        const float dv = d[base + e];
        sa += av * av;
        const float bp = bv - EPS;      sb  += bp * bp;
        const float t1 = av - bv + EPS; sab += t1 * t1;
        const float t2 = av - cv + EPS; sac += t2 * t2;
        const float t3 = av - dv + EPS; sad += t3 * t3;
    }
    #pragma unroll
    for (int m = 16; m >= 1; m >>= 1) {
        sa  += __shfl_xor(sa,  m, 32);
        sb  += __shfl_xor(sb,  m, 32);
        sab += __shfl_xor(sab, m, 32);
        sac += __shfl_xor(sac, m, 32);
        sad += __shfl_xor(sad, m, 32);
    }
    if (lane == 0) {
        na[row] = sa;
        nb[row] = sb;
        const float ap = sqrtf(sab);
        dap[row] = ap;
        const float tc = fmaxf(ap - sqrtf(sac) + MARGIN, 0.f);
        const float td = fmaxf(ap - sqrtf(sad) + MARGIN, 0.f);
        cd[row] = tc + td;
    }
}

// ---------------------------------------------------------------------------
// Kernel 2: fused 512x512x256 fp32 WMMA GEMM + distance + triplet hinge.
// One wave32 per 16x16 output tile; K stepped by 4 through
// V_WMMA_F32_16X16X4_F32 (exact fp32 — matches the fp32 reference).
//
// A fragment (16x4 f32, 2 VGPRs/lane): lanes 0-15 hold M=lane, K=0..1;
// lanes 16-31 hold M=lane-16, K=2..3.  B fragment mirrored over N.
// C/D (16x16 f32, 8 VGPRs): N = lane%16, M = r + 8*(lane>=16).
// ---------------------------------------------------------------------------
__global__ void pairloss_wmma_kernel(const float* __restrict__ A,
                                     const float* __restrict__ Bm,
                                     const float* __restrict__ na,
                                     const float* __restrict__ nb,
                                     const float* __restrict__ dap,
                                     float* __restrict__ tileSums) {
    const int tile = blockIdx.x;          // 1024 tiles (32 x 32)
    const int ti   = tile >> 5;
    const int tj   = tile & 31;
    const int i0   = ti * 16;
    const int j0   = tj * 16;
    const int lane = threadIdx.x & 31;
    const int half = lane >> 4;           // selects K pair 0..1 vs 2..3
    const int l16  = lane & 15;

    const float* __restrict__ arow = A  + (size_t)(i0 + l16) * DIM;
    const float* __restrict__ brow = Bm + (size_t)(j0 + l16) * DIM;

    v8f acc = {0.f, 0.f, 0.f, 0.f, 0.f, 0.f, 0.f, 0.f};

    #pragma unroll 8
    for (int k0 = 0; k0 < DIM; k0 += 4) {
        const int ka = k0 + 2 * half;
        v2f av, bv;
        av.x = arow[ka];
        av.y = arow[ka + 1];
        bv.x = brow[ka]     - EPS;        // fold b' = b - eps into the load
        bv.y = brow[ka + 1] - EPS;
        // D = A x B + C, exact fp32 matrix pipe
        acc = __builtin_amdgcn_wmma_f32_16x16x4_f32(
            /*neg_a=*/false, av, /*neg_b=*/false, bv,
            /*c_mod=*/(short)0, acc, /*reuse_a=*/false, /*reuse_b=*/false);
    }

    // Distance + hinge for the 8 (M,N) elements this lane owns.
    const int   j   = j0 + l16;
    const float nbj = nb[j];
    float s = 0.f;
    #pragma unroll
    for (int r = 0; r < 8; ++r) {
        const int   i    = i0 + r + 8 * half;
        const float d2   = fmaxf(na[i] - 2.0f * acc[r] + nbj, 0.0f);
        const float dist = sqrtf(d2);
        const float t    = dap[i] - dist + MARGIN;
        s += (i != j) ? fmaxf(t, 0.0f) : 0.0f;   // diagonal excluded
    }
    // wave32 butterfly reduction -> one deterministic partial per tile
    #pragma unroll
    for (int m = 16; m >= 1; m >>= 1) s += __shfl_xor(s, m, 32);
    if (lane == 0) tileSums[tile] = s;
}

// ---------------------------------------------------------------------------
// Kernel 3: deterministic final reduction of 1024 tile sums + 512 c/d terms.
// ---------------------------------------------------------------------------
__global__ void finalize_kernel(const float* __restrict__ tileSums,
                                const float* __restrict__ cd,
                                float* __restrict__ out) {
    __shared__ float red[256];
    float s = 0.f;
    for (int t = threadIdx.x; t < 1024; t += 256) s += tileSums[t];
    for (int t = threadIdx.x; t < 512;  t += 256) s += cd[t];
    red[threadIdx.x] = s;
    __syncthreads();
    #pragma unroll
    for (int off = 128; off >= 1; off >>= 1) {
        if (threadIdx.x < (unsigned)off) red[threadIdx.x] += red[threadIdx.x + off];
        __syncthreads();
    }
    if (threadIdx.x == 0) out[0] = red[0] / (513.0f * 512.0f);
}

extern "C" void kernel_launch(void* const* d_in, const int* in_sizes, int n_in,
                              void* d_out, int out_size, void* d_ws, size_t ws_size,
                              hipStream_t stream) {
    (void)in_sizes; (void)n_in; (void)out_size; (void)ws_size;
    const float* a = (const float*)d_in[0];
    const float* b = (const float*)d_in[1];
    const float* c = (const float*)d_in[2];
    const float* d = (const float*)d_in[3];

    float* ws       = (float*)d_ws;       // 3072 floats = 12 KB scratch
    float* na       = ws;                 // [512]
    float* nb       = ws + 512;           // [512]
    float* dap      = ws + 1024;          // [512]
    float* cd       = ws + 1536;          // [512]
    float* tileSums = ws + 2048;          // [1024]

    rowstats_kernel<<<64, 256, 0, stream>>>(a, b, c, d, na, nb, dap, cd);
    pairloss_wmma_kernel<<<1024, 32, 0, stream>>>(a, b, na, nb, dap, tileSums);
    finalize_kernel<<<1, 256, 0, stream>>>(tileSums, cd, (float*)d_out);
}